// Rescaler_54941221650834
// MI455X (gfx1250) — compile-verified
//
#include <hip/hip_runtime.h>

// CDNA5 / gfx1250: wave32, WMMA (not MFMA).
typedef float v2f __attribute__((ext_vector_type(2)));
typedef float v4f __attribute__((ext_vector_type(4)));
typedef float v8f __attribute__((ext_vector_type(8)));

#define IMG_PIX (1024 * 1024)
#define BINS 128
#define NIMG 128

// ---------------------------------------------------------------------------
// 1) zero the per-image histograms (workspace is poisoned / stale each replay)
// ---------------------------------------------------------------------------
__global__ void zero_hist_kernel(unsigned* __restrict__ hist) {
    int i = blockIdx.x * blockDim.x + threadIdx.x;
    if (i < NIMG * BINS) hist[i] = 0u;
}

// ---------------------------------------------------------------------------
// 2) per-image histogram: LDS privatized, 128-bit loads for HBM bandwidth.
//    grid = (64 blocks/image, 128 images), block = 256 threads.
//    Each block covers 16384 pixels = 4096 vec4 (256 thr x 16 iters).
// ---------------------------------------------------------------------------
__global__ void hist_kernel(const float* __restrict__ x,
                            unsigned* __restrict__ ghist) {
    __shared__ unsigned lhist[BINS];
    const int tid = threadIdx.x;
    for (int i = tid; i < BINS; i += blockDim.x) lhist[i] = 0u;
    __syncthreads();

    const int img = blockIdx.y;
    const v4f* xv = (const v4f*)(x + (size_t)img * IMG_PIX) +
                    (size_t)blockIdx.x * 4096;
#pragma unroll
    for (int it = 0; it < 16; ++it) {
        v4f p = xv[it * 256 + tid];
#pragma unroll
        for (int c = 0; c < 4; ++c) {
            float v = p[c];
            // reference: valid = (v>=0 && v<=1); bin = clip(int(v*bins),0,bins-1)
            if (v >= 0.0f && v <= 1.0f) {
                int b = (int)(v * 128.0f);   // truncation == astype(int32)
                if (b > 127) b = 127;        // v==1.0 -> 128 -> clip
                atomicAdd(&lhist[b], 1u);
            }
        }
    }
    __syncthreads();
    for (int i = tid; i < BINS; i += blockDim.x) {
        unsigned cnt = lhist[i];
        if (cnt) atomicAdd(&ghist[img * BINS + i], cnt);
    }
}

// ---------------------------------------------------------------------------
// 3) one MLP layer: OUT(128xN) = act( A(128xK) @ W(KxN) + bias ), fp32 WMMA.
//    V_WMMA_F32_16X16X4_F32 fragments (ISA 7.12.2, wave32):
//      A 16x4 : lane&15 = M row; K = k0 + vgpr + 2*(lane>=16)
//      B 4x16 : lane&15 = N col; K = k0 + vgpr + 2*(lane>=16)
//      C 16x16: lane&15 = N col; M row = vgpr + 8*(lane>=16)
//    Compile-time K/N/flags: the K-loop fully unrolls into a clean
//    load_b64 + v_wmma stream; column masking (only the N=1 layer) is a
//    clamped-address load + v_cndmask select, never an EXEC-divergent branch,
//    so EXEC stays all-ones at every WMMA as the ISA requires.
//    One block of 8 waves; each wave owns independent 16x16 tiles (no syncs).
// ---------------------------------------------------------------------------
template <int K, int N, bool SRCU, bool RELU>
__global__ void mlp_layer_kernel(const void* __restrict__ Asrc,
                                 const float* __restrict__ W,
                                 const float* __restrict__ bias,
                                 float* __restrict__ out) {
    const float*    A  = (const float*)Asrc;
    const unsigned* Au = (const unsigned*)Asrc;

    const unsigned lane   = threadIdx.x & 31u;
    const unsigned wave   = threadIdx.x >> 5;
    const unsigned nWaves = blockDim.x >> 5;
    const int hi = (int)(lane >> 4);    // 0: lanes 0-15, 1: lanes 16-31
    const int lo = (int)(lane & 15u);

    constexpr int ntiles   = (N + 15) / 16;
    constexpr int numTiles = 8 * ntiles;   // M = 128 rows -> 8 M-tiles
    constexpr bool fullN   = (N % 16) == 0;

    for (int t = (int)wave; t < numTiles; t += (int)nWaves) {
        const int mt  = t / ntiles;
        const int nt  = t - mt * ntiles;
        const int row = mt * 16 + lo;              // A-fragment row (< 128)
        const int col = nt * 16 + lo;              // B/C-fragment column
        const bool colOK = fullN || (col < N);
        const int colc = fullN ? col : (colOK ? col : 0);  // safe address

        v8f c = {};
#pragma unroll
        for (int k0 = 0; k0 < K; k0 += 4) {
            const int ka = k0 + 2 * hi;
            v2f a, b;
            if constexpr (SRCU) {
                a.x = (float)Au[row * K + ka];
                a.y = (float)Au[row * K + ka + 1];
            } else {
                a.x = A[row * K + ka];
                a.y = A[row * K + ka + 1];
            }
            const float w0 = W[ka * N + colc];
            const float w1 = W[(ka + 1) * N + colc];
            b.x = colOK ? w0 : 0.0f;   // value select (v_cndmask), no branch
            b.y = colOK ? w1 : 0.0f;
            // 8 args: (neg_a, A, neg_b, B, c_mod, C, reuse_a, reuse_b)
            c = __builtin_amdgcn_wmma_f32_16x16x4_f32(
                    false, a, false, b, (short)0, c, false, false);
        }
        if (colOK) {
            const float bb = bias[colc];
#pragma unroll
            for (int r = 0; r < 8; ++r) {
                const int orow = mt * 16 + r + 8 * hi;
                float v = c[r] + bb;
                if constexpr (RELU) v = fmaxf(v, 0.0f);
                out[orow * N + col] = v;
            }
        }
    }
}

// ---------------------------------------------------------------------------
// 4) out = x * w[image]; pure bandwidth pass (the roofline-dominant kernel:
//    1 GB of traffic at 23.3 TB/s). Non-temporal 128-bit in/out — the stream
//    is 512 MB each way with zero reuse, so don't thrash the 192 MB L2.
//    grid = (1024, 128), block = 256: 1024*256 vec4 = 1M pix per image.
// ---------------------------------------------------------------------------
__global__ void scale_kernel(const float* __restrict__ x,
                             const float* __restrict__ w,
                             float* __restrict__ out) {
    const int img = blockIdx.y;
    const float s = w[img];  // wave-uniform -> scalar load
    const size_t i = (size_t)img * (IMG_PIX / 4) +
                     (size_t)blockIdx.x * blockDim.x + threadIdx.x;
    const v4f p = __builtin_nontemporal_load((const v4f*)x + i);
    const v4f o = p * s;
    __builtin_nontemporal_store(o, (v4f*)out + i);
}

// ---------------------------------------------------------------------------
// launcher
// inputs: x, W1, b1, W2, b2, W3, b3, W4, b4, bins(=128)
// workspace: [0,64K) u32 hist | act1 16K | act2 32K | act3 64K | w 512B
// ---------------------------------------------------------------------------
extern "C" void kernel_launch(void* const* d_in, const int* in_sizes, int n_in,
                              void* d_out, int out_size, void* d_ws, size_t ws_size,
                              hipStream_t stream) {
    const float* x  = (const float*)d_in[0];
    const float* W1 = (const float*)d_in[1];
    const float* b1 = (const float*)d_in[2];
    const float* W2 = (const float*)d_in[3];
    const float* b2 = (const float*)d_in[4];
    const float* W3 = (const float*)d_in[5];
    const float* b3 = (const float*)d_in[6];
    const float* W4 = (const float*)d_in[7];
    const float* b4 = (const float*)d_in[8];
    // d_in[9] = bins = 128 (fixed by the reference setup)

    char* ws = (char*)d_ws;
    unsigned* hist = (unsigned*)ws;                              // 64 KB
    float* act1 = (float*)(ws + 65536);                          // 128x32
    float* act2 = (float*)(ws + 65536 + 16384);                  // 128x64
    float* act3 = (float*)(ws + 65536 + 16384 + 32768);          // 128x128
    float* wv   = (float*)(ws + 65536 + 16384 + 32768 + 65536);  // 128

    zero_hist_kernel<<<64, 256, 0, stream>>>(hist);
    hist_kernel<<<dim3(64, NIMG), 256, 0, stream>>>(x, hist);

    mlp_layer_kernel<128,  32, true,  true ><<<1, 256, 0, stream>>>(hist, W1, b1, act1);
    mlp_layer_kernel< 32,  64, false, true ><<<1, 256, 0, stream>>>(act1, W2, b2, act2);
    mlp_layer_kernel< 64, 128, false, true ><<<1, 256, 0, stream>>>(act2, W3, b3, act3);
    mlp_layer_kernel<128,   1, false, false><<<1, 256, 0, stream>>>(act3, W4, b4, wv);

    scale_kernel<<<dim3(1024, NIMG), 256, 0, stream>>>(x, wv, (float*)d_out);
}